// AELNGCN_53635551592932
// MI455X (gfx1250) — compile-verified
//
#include <hip/hip_runtime.h>
#include <hip/hip_bf16.h>

typedef float v2f __attribute__((ext_vector_type(2)));
typedef float v8f __attribute__((ext_vector_type(8)));
typedef int   v4i __attribute__((ext_vector_type(4)));

// async global->LDS builtin wants: (as1 v4i*, as3 v4i*, imm offset, imm cpol)
#define GPTR4(p) ((__attribute__((address_space(1))) v4i*)(p))
#define LPTR4(p) ((__attribute__((address_space(3))) v4i*)(p))

__device__ __forceinline__ float sigmoidf_(float x) {
    return 1.0f / (1.0f + __expf(-x));
}

__device__ __forceinline__ void atomic_add_f32(float* p, float v) {
    __hip_atomic_fetch_add(p, v, __ATOMIC_RELAXED, __HIP_MEMORY_SCOPE_AGENT);
}

// ---------------- zero ----------------
__global__ void zero_kernel(float* __restrict__ p, long long n) {
    long long i = (long long)blockIdx.x * blockDim.x + threadIdx.x;
    long long stride = (long long)gridDim.x * blockDim.x;
    for (; i < n; i += stride) p[i] = 0.0f;
}

// ---------------- edge MLP: ew = sigmoid(relu(in@E1 + e1b)@E2 + e2b) ----------------
// one wave per 16 edges; two 16-col tiles of the 32 hidden units via WMMA f32 16x16x4
__global__ void edge_mlp_kernel(const float* __restrict__ in,   // [E,64]
                                const float* __restrict__ E1,   // [64,32]
                                const float* __restrict__ e1b,  // [32]
                                const float* __restrict__ E2,   // [32,1]
                                const float* __restrict__ e2b,  // [1]
                                float* __restrict__ ew, int E) {
    int lane = threadIdx.x & 31;
    int wave = threadIdx.x >> 5;
    int wtile = blockIdx.x * (blockDim.x >> 5) + wave;
    int e0 = wtile * 16;
    if (e0 >= E) return;

    int m  = lane & 15;          // row (edge) for A / col for B,C
    int kb = (lane >> 4) * 2;    // K sub-offset per half-wave
    int e  = e0 + m;
    if (e >= E) e = E - 1;       // clamp: branch-free, duplicate rows are never stored
    const float* arow = in + (long long)e * 64;

    v8f c0 = {}; v8f c1 = {};
    #pragma unroll
    for (int kk = 0; kk < 64; kk += 4) {
        int k0 = kk + kb, k1 = kk + kb + 1;
        v2f a = *(const v2f*)(arow + k0);          // contiguous -> b64 load
        v2f b0, b1;
        b0.x = E1[k0 * 32 + m];      b0.y = E1[k1 * 32 + m];
        b1.x = E1[k0 * 32 + 16 + m]; b1.y = E1[k1 * 32 + 16 + m];
        c0 = __builtin_amdgcn_wmma_f32_16x16x4_f32(false, a, false, b0, (short)0, c0, false, false);
        c1 = __builtin_amdgcn_wmma_f32_16x16x4_f32(false, a, false, b1, (short)0, c1, false, false);
    }

    float bias0 = e1b[m], bias1 = e1b[16 + m];
    float w0 = E2[m], w1 = E2[16 + m];
    float eb = e2b[0];

    float s[8];
    #pragma unroll
    for (int r = 0; r < 8; ++r)
        s[r] = fmaxf(c0[r] + bias0, 0.0f) * w0 + fmaxf(c1[r] + bias1, 0.0f) * w1;

    // reduce across the 16 lanes of each half-wave (cols of h)
    #pragma unroll
    for (int r = 0; r < 8; ++r) {
        #pragma unroll
        for (int off = 1; off < 16; off <<= 1)
            s[r] += __shfl_xor(s[r], off, 32);
    }

    if (m == 0) {
        int rbase = (lane >> 4) * 8;   // half 0 -> rows 0..7, half 1 -> rows 8..15
        #pragma unroll
        for (int r = 0; r < 8; ++r) {
            int ee = e0 + rbase + r;
            if (ee < E) ew[ee] = sigmoidf_(s[r] + eb);
        }
    }
}

// ---------------- deg[src] += ew ----------------
__global__ void deg_kernel(const float* __restrict__ ew, const int* __restrict__ ei,
                           float* __restrict__ deg, int E) {
    int e = blockIdx.x * blockDim.x + threadIdx.x;
    if (e < E) atomic_add_f32(&deg[ei[e]], ew[e]);
}

// ---------------- dinv in place over deg ----------------
__global__ void dinv_kernel(float* __restrict__ deg, int N) {
    int n = blockIdx.x * blockDim.x + threadIdx.x;
    if (n < N) {
        float d = deg[n];
        deg[n] = (d > 0.0f) ? rsqrtf(fmaxf(d, 1e-12f)) : 0.0f;
    }
}

// ---------------- nw in place over ew ----------------
__global__ void nw_kernel(float* __restrict__ ew, const int* __restrict__ ei,
                          const float* __restrict__ dinv, int E) {
    int e = blockIdx.x * blockDim.x + threadIdx.x;
    if (e < E) {
        int s = ei[e], d = ei[E + e];
        ew[e] = -dinv[s] * ew[e] * dinv[d];
    }
}

// ---------------- SpMM: y[dst] += nw * x[src]; one wave per edge ----------------
__global__ void spmm_kernel(const float* __restrict__ x, const float* __restrict__ nw,
                            const int* __restrict__ ei, float* __restrict__ y, int E) {
    int lane = threadIdx.x & 31;
    int wave = threadIdx.x >> 5;
    int e = blockIdx.x * (blockDim.x >> 5) + wave;
    if (e >= E) return;
    int s = ei[e], d = ei[E + e];
    float w = nw[e];
    float4 v = ((const float4*)(x + (long long)s * 128))[lane];
    float* yd = y + (long long)d * 128 + lane * 4;
    atomic_add_f32(yd + 0, w * v.x);
    atomic_add_f32(yd + 1, w * v.y);
    atomic_add_f32(yd + 2, w * v.z);
    atomic_add_f32(yd + 3, w * v.w);
}

// ---------------- t2 = 2*t2 - xin ----------------
__global__ void t2_fix_kernel(float* __restrict__ t2, const float* __restrict__ xin, long long n) {
    long long i = (long long)blockIdx.x * blockDim.x + threadIdx.x;
    long long stride = (long long)gridDim.x * blockDim.x;
    for (; i < n; i += stride) t2[i] = 2.0f * t2[i] - xin[i];
}

// ---------------- out = relu(x0@W0 + x1@W1 + x2@W2 + b) [+ res] ----------------
// one wave per 16x16 output tile; 8 waves/block cover all 128 cols of one row tile.
// A tiles (3 x 16x128) staged to LDS once per block via async global->LDS copies.
__global__ void cheb_gemm_kernel(const float* __restrict__ x0, const float* __restrict__ x1,
                                 const float* __restrict__ x2,
                                 const float* __restrict__ W,    // [3,128,128]
                                 const float* __restrict__ bias, // [128]
                                 float* __restrict__ out, const float* __restrict__ res,
                                 int N) {
    __shared__ float lds[3 * 16 * 128];    // 24 KB

    int lane = threadIdx.x & 31;
    int wave = threadIdx.x >> 5;
    int m0 = blockIdx.x * 16;

    // ---- cooperative A-tile staging (1536 float4 total; 6 per thread) ----
    {
        #pragma unroll
        for (int i = threadIdx.x; i < 1536; i += 256) {
            int mat = i >> 9;              // 0..2   (512 float4 per 16x128 tile)
            int rem = i & 511;
            int r   = rem >> 5;            // 0..15
            int c4  = (rem & 31) << 2;     // 0,4,...,124
            int row = m0 + r; if (row >= N) row = N - 1;
            const float* src = (mat == 0 ? x0 : (mat == 1 ? x1 : x2)) + (long long)row * 128 + c4;
            float* dst = &lds[mat * 2048 + r * 128 + c4];
#if __has_builtin(__builtin_amdgcn_global_load_async_to_lds_b128)
            __builtin_amdgcn_global_load_async_to_lds_b128(GPTR4(src), LPTR4(dst), 0, 0);
#else
            *(float4*)dst = *(const float4*)src;
#endif
        }
    }
#if __has_builtin(__builtin_amdgcn_global_load_async_to_lds_b128) && __has_builtin(__builtin_amdgcn_s_wait_asynccnt)
    __builtin_amdgcn_s_wait_asynccnt(0);
#endif
    __syncthreads();

    int n0 = wave * 16;
    int nn = lane & 15;
    int kb = (lane >> 4) * 2;

    const float* W0 = W;
    const float* W1 = W + 128 * 128;
    const float* W2 = W + 2 * 128 * 128;

    v8f c0 = {}; v8f c1 = {}; v8f c2 = {};
    #pragma unroll 4
    for (int kk = 0; kk < 128; kk += 4) {
        int k0 = kk + kb, k1 = kk + kb + 1;
        v2f a0 = *(const v2f*)&lds[       nn * 128 + k0];   // ds_load_b64
        v2f a1 = *(const v2f*)&lds[2048 + nn * 128 + k0];
        v2f a2 = *(const v2f*)&lds[4096 + nn * 128 + k0];
        v2f b0, b1, b2;
        b0.x = W0[k0 * 128 + n0 + nn]; b0.y = W0[k1 * 128 + n0 + nn];
        b1.x = W1[k0 * 128 + n0 + nn]; b1.y = W1[k1 * 128 + n0 + nn];
        b2.x = W2[k0 * 128 + n0 + nn]; b2.y = W2[k1 * 128 + n0 + nn];
        c0 = __builtin_amdgcn_wmma_f32_16x16x4_f32(false, a0, false, b0, (short)0, c0, false, false);
        c1 = __builtin_amdgcn_wmma_f32_16x16x4_f32(false, a1, false, b1, (short)0, c1, false, false);
        c2 = __builtin_amdgcn_wmma_f32_16x16x4_f32(false, a2, false, b2, (short)0, c2, false, false);
    }

    float bv = bias[n0 + nn];
    #pragma unroll
    for (int r = 0; r < 8; ++r) {
        int orow = m0 + r + ((lane >> 4) << 3);
        if (orow < N) {
            long long oi = (long long)orow * 128 + n0 + nn;
            float v = fmaxf(c0[r] + c1[r] + c2[r] + bv, 0.0f);
            if (res) v += res[oi];
            out[oi] = v;
        }
    }
}

extern "C" void kernel_launch(void* const* d_in, const int* in_sizes, int n_in,
                              void* d_out, int out_size, void* d_ws, size_t ws_size,
                              hipStream_t stream) {
    const float* features = (const float*)d_in[0];
    const float* edgenet  = (const float*)d_in[1];
    const float* E1  = (const float*)d_in[2];
    const float* e1b = (const float*)d_in[3];
    const float* E2  = (const float*)d_in[4];
    const float* e2b = (const float*)d_in[5];
    const float* W   = (const float*)d_in[6];
    const float* b   = (const float*)d_in[7];
    const int*   ei  = (const int*)d_in[8];

    int N = in_sizes[0] / 128;
    int E = in_sizes[1] / 64;
    int NGL = in_sizes[7] / 128;
    long long NF = (long long)N * 128;

    float* out = (float*)d_out;
    float* ws  = (float*)d_ws;
    float* ew  = ws;             // [E]   ew -> nw (in place)
    float* deg = ew + E;         // [N]   deg -> dinv (in place)
    float* t1  = deg + N;        // [N*128]
    float* t2  = t1 + NF;        // [N*128]
    float* xa  = t2 + NF;        // [N*128]
    float* xb  = out;            // reuse output buffer as ping-pong partner

    // --- edge network + normalization ---
    int mlpBlocks = ((E + 15) / 16 + 7) / 8;                 // 8 waves/block, 16 edges/wave
    edge_mlp_kernel<<<mlpBlocks, 256, 0, stream>>>(edgenet, E1, e1b, E2, e2b, ew, E);
    zero_kernel<<<256, 256, 0, stream>>>(deg, N);
    deg_kernel<<<(E + 255) / 256, 256, 0, stream>>>(ew, ei, deg, E);
    dinv_kernel<<<(N + 255) / 256, 256, 0, stream>>>(deg, N);
    nw_kernel<<<(E + 255) / 256, 256, 0, stream>>>(ew, ei, deg, E);
    const float* nw = ew;

    int spmmBlocks = (E + 7) / 8;          // 8 waves/block, 1 edge/wave
    int gemmBlocks = (N + 15) / 16;        // 1 row tile/block

    float* cur = nullptr;                  // layer output buffer
    const float* xin;
    for (int l = 0; l < NGL; ++l) {
        if (l == 0) xin = features;
        else        xin = cur;
        float* xout = (l == 0) ? xa : ((cur == xa) ? xb : xa);
        if (l == NGL - 1) xout = out;      // final layer writes the real output
        const float* res = (l >= 1 && l <= NGL - 2) ? xin : nullptr;

        zero_kernel<<<2048, 256, 0, stream>>>(t1, NF);
        spmm_kernel<<<spmmBlocks, 256, 0, stream>>>(xin, nw, ei, t1, E);
        zero_kernel<<<2048, 256, 0, stream>>>(t2, NF);
        spmm_kernel<<<spmmBlocks, 256, 0, stream>>>(t1, nw, ei, t2, E);
        t2_fix_kernel<<<2048, 256, 0, stream>>>(t2, xin, NF);
        cheb_gemm_kernel<<<gemmBlocks, 256, 0, stream>>>(
            xin, t1, t2, W + (long long)l * 3 * 128 * 128, b + l * 128, xout, res, N);
        cur = xout;
    }
    if (cur != out) {
        (void)hipMemcpyAsync(out, cur, NF * sizeof(float), hipMemcpyDeviceToDevice, stream);
    }
}